// DetectionLoss_16801912062786
// MI455X (gfx1250) — compile-verified
//
#include <hip/hip_runtime.h>

#define B_ 1024
#define S_ 13
#define A_ 5
#define C_ 20
#define E_ 25            // 5 + C
#define CH_ (A_ * E_)    // 125
#define CELLS_ (S_ * S_) // 169
#define NCELL_ (B_ * CELLS_)
#define DX_ 32.0f        // 416 / 13
#define IMGW_ 416.0f

typedef float v2f __attribute__((ext_vector_type(2)));
typedef float v8f __attribute__((ext_vector_type(8)));

// ---------------------------------------------------------------------------
// Stage 1: one thread per (b, sy, sx) cell. Cell-minor thread order keeps the
// channel-strided pred loads coalesced within a wave. Non-obj cells touch only
// the 5 objectness channels; obj cells touch 20 box params + 21 best-anchor
// values. Block-level LDS tree reduction -> one partial per block in d_ws.
// ---------------------------------------------------------------------------
__global__ __launch_bounds__(256) void yolo_loss_stage1(
    const float* __restrict__ pred,
    const float* __restrict__ yhat,
    const float* __restrict__ prior,   // (A,2)
    float* __restrict__ partial)
{
  __shared__ float red[256];
  const int tid = threadIdx.x;
  const int n = blockIdx.x * 256 + tid;
  float acc = 0.0f;

  if (n < NCELL_) {
    const int b    = n / CELLS_;
    const int cell = n - b * CELLS_;
    const int iy   = cell / S_;
    const int jx   = cell - iy * S_;

    const float* yh = yhat + (size_t)n * 6;
    const float g_obj = yh[0], g_tx = yh[1], g_ty = yh[2];
    const float g_tw = yh[3], g_th = yh[4], g_cls = yh[5];

    const float* pb = pred + (size_t)b * (CH_ * CELLS_) + cell;
    __builtin_prefetch(pb, 0, 3);   // global_prefetch_b8

    if (g_obj != 1.0f) {
      // non-object: LAMBDA_NONOBJ * sum_a p_obj^2
      float s = 0.0f;
#pragma unroll
      for (int a = 0; a < A_; ++a) {
        const float po = pb[(a * E_ + 0) * CELLS_];
        s += po * po;
      }
      acc = 0.5f * s;
    } else {
      // ground-truth box (corners)
      const float g_bx = DX_ * (float)jx + DX_ * g_tx;
      const float g_by = DX_ * (float)iy + DX_ * g_ty;
      const float g_bw = g_tw * IMGW_;
      const float g_bh = g_th * IMGW_;
      const float gx1 = g_bx - 0.5f * g_bw, gx2 = g_bx + 0.5f * g_bw;
      const float gy1 = g_by - 0.5f * g_bh, gy2 = g_by + 0.5f * g_bh;
      const float garea = g_bw * g_bh;

      float best_iou = 0.0f;
      int   best_a = 0;
      float btx = 0.f, bty = 0.f, btw = 0.f, bth = 0.f;

#pragma unroll
      for (int a = 0; a < A_; ++a) {
        const float ptx = pb[(a * E_ + 1) * CELLS_];
        const float pty = pb[(a * E_ + 2) * CELLS_];
        const float ptw = pb[(a * E_ + 3) * CELLS_];
        const float pth = pb[(a * E_ + 4) * CELLS_];

        const float p_bx = DX_ * (float)jx + DX_ * ptx;
        const float p_by = DX_ * (float)iy + DX_ * pty;
        const float p_bw = prior[2 * a + 0] * ptw * IMGW_;
        const float p_bh = prior[2 * a + 1] * pth * IMGW_;

        const float px1 = p_bx - 0.5f * p_bw, px2 = p_bx + 0.5f * p_bw;
        const float py1 = p_by - 0.5f * p_bh, py2 = p_by + 0.5f * p_bh;

        float iw = fminf(px2, gx2) - fmaxf(px1, gx1); iw = fmaxf(iw, 0.0f);
        float ih = fminf(py2, gy2) - fmaxf(py1, gy1); ih = fmaxf(ih, 0.0f);
        const float inter = iw * ih;
        const float uni   = p_bw * p_bh + garea - inter;
        const float iou   = inter / (uni + 1e-6f);

        const bool take = (a == 0) || (iou > best_iou);  // first-max argmax
        if (take) { best_iou = iou; best_a = a; btx = ptx; bty = pty; btw = ptw; bth = pth; }
      }

      const float pobj = pb[(best_a * E_ + 0) * CELLS_];
      const float dtx = btx - g_tx, dty = bty - g_ty;
      const float dtw = btw - g_tw, dth = bth - g_th;
      const float coord = dtx * dtx + dty * dty + dtw * dtw + dth * dth;
      const float dobj = pobj - best_iou;

      const int gc = (int)(g_cls + 0.5f);  // g_cls is an exact integer
      float ce = 0.0f;
      const float* pc = pb + (best_a * E_ + 5) * CELLS_;
#pragma unroll
      for (int c = 0; c < C_; ++c) {
        const float d = pc[c * CELLS_] - ((c == gc) ? 1.0f : 0.0f);
        ce += d * d;
      }
      acc = 5.0f * (coord + dobj * dobj + ce);
    }
  }

  red[tid] = acc;
  __syncthreads();
#pragma unroll
  for (int s = 128; s > 0; s >>= 1) {
    if (tid < s) red[tid] += red[tid + s];
    __syncthreads();
  }
  if (tid == 0) partial[blockIdx.x] = red[0];
}

// ---------------------------------------------------------------------------
// Stage 2: single wave32. Each lane folds the block partials into 2 floats ->
// 64 values packed into the B operand of V_WMMA_F32_16X16X4_F32 with an
// all-ones A matrix, so D = per-column sums regardless of operand lane layout.
// Summing the 16 column sums (D row 0, lanes 0..15) gives the grand total.
// ---------------------------------------------------------------------------
__global__ __launch_bounds__(32) void yolo_loss_stage2(
    const float* __restrict__ partial, int np, float* __restrict__ out)
{
  const int lane = threadIdx.x;

  v2f bmat; bmat.x = 0.0f; bmat.y = 0.0f;
  for (int i = lane; i < np; i += 64)       bmat.x += partial[i];
  for (int i = lane + 32; i < np; i += 64)  bmat.y += partial[i];

  v2f amat; amat.x = 1.0f; amat.y = 1.0f;   // all-ones A (16x4)
  v8f cmat = {};
  // D = A * B + 0 : every element of D row m is a column sum of B.
  cmat = __builtin_amdgcn_wmma_f32_16x16x4_f32(
      /*neg_a=*/false, amat, /*neg_b=*/false, bmat,
      /*c_mod=*/(short)0, cmat, /*reuse_a=*/false, /*reuse_b=*/false);

  __shared__ float tmp[32];
  tmp[lane] = cmat[0];   // lanes 0..15: D[0][0..15] (column sums)
  __syncthreads();

  if (lane == 0) {
    float s = 0.0f;
#pragma unroll
    for (int k = 0; k < 16; ++k) s += tmp[k];
    out[0] = s / (float)B_;
  }
}

extern "C" void kernel_launch(void* const* d_in, const int* in_sizes, int n_in,
                              void* d_out, int out_size, void* d_ws, size_t ws_size,
                              hipStream_t stream) {
  const float* pred  = (const float*)d_in[0];
  const float* yhat  = (const float*)d_in[1];
  const float* prior = (const float*)d_in[2];
  float* partial = (float*)d_ws;

  const int nblocks = (NCELL_ + 255) / 256;  // 676 (exact)
  yolo_loss_stage1<<<nblocks, 256, 0, stream>>>(pred, yhat, prior, partial);
  yolo_loss_stage2<<<1, 32, 0, stream>>>(partial, nblocks, (float*)d_out);
}